// SimpleConvLSTM_16389595201595
// MI455X (gfx1250) — compile-verified
//
#include <hip/hip_runtime.h>
#include <hip/hip_bf16.h>

typedef __attribute__((ext_vector_type(2))) float v2f;
typedef __attribute__((ext_vector_type(8))) float v8f;

#define Bn   8
#define Tn   16
#define Hn   112
#define Wn   112
#define Cn   3
#define Fn   16
#define HPn  56
#define WPn  56
#define HOn  54
#define WOn  54
#define NCn  6
#define KTOT 144            // 3*3*16 im2col depth, = 36 * 4
#define NG   64             // 4F gate channels

__device__ __forceinline__ float hsig(float x) {
    return fminf(fmaxf(0.2f * x + 0.5f, 0.f), 1.f);
}

__global__ void zero_kernel(float* p, int n) {
    int i = blockIdx.x * blockDim.x + threadIdx.x;
    if (i < n) p[i] = 0.f;
}

// Per-frame 7x7 SAME conv (C=3 -> F=16) + ReLU + 2x2 maxpool.
// One thread per pooled pixel (b,hp,wp); computes all 16 channels.
__global__ void conv_pool_kernel(const float* __restrict__ x,
                                 const float* __restrict__ w,
                                 const float* __restrict__ bias,
                                 float* __restrict__ pooled, int t) {
    int idx = blockIdx.x * blockDim.x + threadIdx.x;
    if (idx >= Bn * HPn * WPn) return;
    int wp = idx % WPn; int tmp = idx / WPn;
    int hp = tmp % HPn; int b = tmp / HPn;
    const float* xf = x + ((size_t)(b * Tn + t)) * Hn * Wn * Cn;
    float best[Fn];
    #pragma unroll
    for (int f = 0; f < Fn; ++f) best[f] = 0.f;   // relu output >= 0
    for (int dy = 0; dy < 2; ++dy)
    for (int dx = 0; dx < 2; ++dx) {
        float acc[Fn];
        #pragma unroll
        for (int f = 0; f < Fn; ++f) acc[f] = bias[f];
        int y0 = 2 * hp + dy - 3, x0 = 2 * wp + dx - 3;
        for (int ky = 0; ky < 7; ++ky) {
            int yi = y0 + ky; if (yi < 0 || yi >= Hn) continue;
            for (int kx = 0; kx < 7; ++kx) {
                int xi = x0 + kx; if (xi < 0 || xi >= Wn) continue;
                const float* xp = xf + ((size_t)yi * Wn + xi) * Cn;
                const float* wk = w + ((ky * 7 + kx) * Cn) * Fn;
                #pragma unroll
                for (int ci = 0; ci < Cn; ++ci) {
                    float xv = xp[ci];
                    const float* wc = wk + ci * Fn;
                    #pragma unroll
                    for (int f = 0; f < Fn; ++f) acc[f] += xv * wc[f];
                }
            }
        }
        #pragma unroll
        for (int f = 0; f < Fn; ++f)
            best[f] = fmaxf(best[f], fmaxf(acc[f], 0.f));
    }
    float* op = pooled + (size_t)idx * Fn;
    #pragma unroll
    for (int f = 0; f < Fn; ++f) op[f] = best[f];
}

// 256-thread block = 8 wave32 waves. Block cooperatively stages both gate
// weight tensors (2 x 144 x 64 f32 = 73.7 KB) into WGP LDS, repacked so
// each B fragment (K-pair) is one ds_load_b64. Each wave then runs an
// independent im2col GEMM tile: M = 16 output columns along Wo,
// N = 64 gates (4 WMMA tiles), K = 144 (36 steps of V_WMMA_F32_16X16X4_F32)
// for the VALID input conv plus 36 steps for the SAME recurrent conv,
// followed by per-lane gate activations and the h/c state update.
// Grid covers exactly 54 rows * 4 x-tiles = 216 tiles per image: 27 blocks
// of 8 waves, no idle waves.
__global__ void __launch_bounds__(256)
lstm_step_kernel(const float* __restrict__ xt,   // [B,56,56,16]
                 const float* __restrict__ wk,   // [3,3,16,64] -> flat k*64+n
                 const float* __restrict__ wr,   // [3,3,16,64]
                 const float* __restrict__ bias, // [64]
                 const float* __restrict__ h_in, // [B,54,54,16]
                 float* __restrict__ h_out,      // [B,54,54,16]
                 float* __restrict__ c_buf) {    // [B,54,54,16]
    extern __shared__ float lws[];               // 2 * KTOT * NG floats
    float* lwk = lws;                            // repacked input-conv weights
    float* lwr = lws + KTOT * NG;                // repacked recurrent weights

    // Stage + repack: pair layout d = (k>>1)*128 + n*2 + (k&1) so that the
    // two K values of a WMMA B fragment are adjacent (one ds_load_b64).
    for (int i = threadIdx.x; i < KTOT * NG; i += 256) {
        int k = i >> 6, n = i & 63;
        int d = ((k >> 1) * NG + n) * 2 + (k & 1);
        lwk[d] = wk[i];
        lwr[d] = wr[i];
    }
    __syncthreads();

    const int lane = threadIdx.x & 31;
    const int wave = threadIdx.x >> 5;
    const int half = lane >> 4;      // K-half for A/B fragments
    const int lN   = lane & 15;      // A: row M; B,C,D: column N
    const int b    = blockIdx.y;
    const int ti   = blockIdx.x * 8 + wave;       // 0..215
    const int yo   = ti >> 2;
    const int xob  = (ti & 3) * 16;
    const int xo   = xob + lN;                    // A-matrix row position
    const int xc   = xo < WOn ? xo : WOn - 1;     // clamp for safe addressing

    // Pull the recurrent-state row toward the WGP while the zx GEMM runs.
    __builtin_prefetch(&h_in[(((size_t)(b * HOn) + yo) * WOn + xc) * Fn], 0, 3);

    v8f acc[4];
    #pragma unroll
    for (int nt = 0; nt < 4; ++nt) {
        float bv = bias[nt * 16 + lN];   // same N for all 8 rows of this lane
        v8f tv;
        #pragma unroll
        for (int v = 0; v < 8; ++v) tv[v] = bv;
        acc[nt] = tv;
    }

    // ---- zx: VALID 3x3 conv on pooled input, K = 144 ----
    for (int ks = 0; ks < 36; ++ks) {
        int k0 = ks * 4;
        v2f a;
        #pragma unroll
        for (int j = 0; j < 2; ++j) {
            int k  = k0 + 2 * half + j;
            int ci = k & 15, kxy = k >> 4;
            int ky = kxy / 3, kx = kxy - 3 * ky;
            a[j] = xt[(((size_t)(b * HPn) + (yo + ky)) * WPn + (xc + kx)) * Fn + ci];
        }
        int pbase = (2 * ks + half) * (2 * NG);   // pair row in LDS
        #pragma unroll
        for (int nt = 0; nt < 4; ++nt) {
            v2f bb = *(const v2f*)&lwk[pbase + (nt * 16 + lN) * 2];
            acc[nt] = __builtin_amdgcn_wmma_f32_16x16x4_f32(
                false, a, false, bb, (short)0, acc[nt], false, false);
        }
    }

    // ---- zh: SAME 3x3 conv on h_in (zero padded), K = 144 ----
    for (int ks = 0; ks < 36; ++ks) {
        int k0 = ks * 4;
        v2f a;
        #pragma unroll
        for (int j = 0; j < 2; ++j) {
            int k  = k0 + 2 * half + j;
            int ci = k & 15, kxy = k >> 4;
            int ky = kxy / 3, kx = kxy - 3 * ky;
            int yi = yo + ky - 1;
            int xi = xc + kx - 1;
            float v = 0.f;
            if (yi >= 0 && yi < HOn && xi >= 0 && xi < WOn)
                v = h_in[(((size_t)(b * HOn) + yi) * WOn + xi) * Fn + ci];
            a[j] = v;
        }
        int pbase = (2 * ks + half) * (2 * NG);
        #pragma unroll
        for (int nt = 0; nt < 4; ++nt) {
            v2f bb = *(const v2f*)&lwr[pbase + (nt * 16 + lN) * 2];
            acc[nt] = __builtin_amdgcn_wmma_f32_16x16x4_f32(
                false, a, false, bb, (short)0, acc[nt], false, false);
        }
    }

    // ---- gates + state update (C/D layout: lane n = lN, row m = v + 8*half) ----
    #pragma unroll
    for (int v = 0; v < 8; ++v) {
        int m   = v + 8 * half;
        int xow = xob + m;
        int ok  = xow < WOn;
        int xcw = ok ? xow : WOn - 1;
        size_t idx = (((size_t)(b * HOn) + yo) * WOn + xcw) * Fn + lN;
        float c_old = c_buf[idx];
        float zi = acc[0][v], zf = acc[1][v], zg = acc[2][v], zo = acc[3][v];
        float ig = hsig(zi), fg = hsig(zf), og = hsig(zo);
        float c_new = fg * c_old + ig * tanhf(zg);
        float h_new = og * tanhf(c_new);
        if (ok) {
            c_buf[idx] = c_new;
            h_out[idx] = h_new;
        }
    }
}

// Global average pool over 54*54, dense 16->6, softmax. One block per batch.
__global__ void pool_dense_softmax_kernel(const float* __restrict__ h,
                                          const float* __restrict__ dw,
                                          const float* __restrict__ db,
                                          float* __restrict__ out) {
    __shared__ float red[256 * Fn];
    int b = blockIdx.x, tid = threadIdx.x;
    float local[Fn];
    #pragma unroll
    for (int f = 0; f < Fn; ++f) local[f] = 0.f;
    for (int p = tid; p < HOn * WOn; p += 256) {
        const float* hp = h + ((size_t)b * HOn * WOn + p) * Fn;
        #pragma unroll
        for (int f = 0; f < Fn; ++f) local[f] += hp[f];
    }
    #pragma unroll
    for (int f = 0; f < Fn; ++f) red[tid * Fn + f] = local[f];
    __syncthreads();
    for (int s = 128; s > 0; s >>= 1) {
        if (tid < s)
            for (int f = 0; f < Fn; ++f)
                red[tid * Fn + f] += red[(tid + s) * Fn + f];
        __syncthreads();
    }
    if (tid == 0) {
        float logits[NCn];
        float mx = -1e30f;
        for (int j = 0; j < NCn; ++j) {
            float s = db[j];
            for (int f = 0; f < Fn; ++f)
                s += (red[f] / (float)(HOn * WOn)) * dw[f * NCn + j];
            logits[j] = s;
            mx = fmaxf(mx, s);
        }
        float se = 0.f;
        for (int j = 0; j < NCn; ++j) { logits[j] = __expf(logits[j] - mx); se += logits[j]; }
        for (int j = 0; j < NCn; ++j) out[b * NCn + j] = logits[j] / se;
    }
}

extern "C" void kernel_launch(void* const* d_in, const int* in_sizes, int n_in,
                              void* d_out, int out_size, void* d_ws, size_t ws_size,
                              hipStream_t stream) {
    const float* x      = (const float*)d_in[0];
    const float* conv_w = (const float*)d_in[1];
    const float* conv_b = (const float*)d_in[2];
    const float* lk     = (const float*)d_in[3];
    const float* lr     = (const float*)d_in[4];
    const float* lb     = (const float*)d_in[5];
    const float* dw     = (const float*)d_in[6];
    const float* db     = (const float*)d_in[7];
    float* out = (float*)d_out;
    float* ws  = (float*)d_ws;

    const size_t pooled_sz = (size_t)Bn * HPn * WPn * Fn;  // 401,408 floats
    const size_t hc_sz     = (size_t)Bn * HOn * WOn * Fn;  // 373,248 floats
    float* pooled = ws;
    float* h_a    = pooled + pooled_sz;
    float* h_b    = h_a + hc_sz;
    float* c      = h_b + hc_sz;

    // zero h_a, h_b, c (contiguous)
    int zn = (int)(hc_sz * 3);
    zero_kernel<<<(zn + 255) / 256, 256, 0, stream>>>(h_a, zn);

    const size_t lds_bytes = (size_t)2 * KTOT * NG * sizeof(float);  // 73,728 B
    int cpN = Bn * HPn * WPn;
    for (int t = 0; t < Tn; ++t) {
        conv_pool_kernel<<<(cpN + 255) / 256, 256, 0, stream>>>(
            x, conv_w, conv_b, pooled, t);
        float* hin  = (t & 1) ? h_b : h_a;
        float* hout = (t & 1) ? h_a : h_b;
        // 216 tiles per image = 27 blocks * 8 waves
        lstm_step_kernel<<<dim3(27, Bn), 256, lds_bytes, stream>>>(
            pooled, lk, lr, lb, hin, hout, c);
    }
    // t=15 (odd) wrote h_a
    pool_dense_softmax_kernel<<<Bn, 256, 0, stream>>>(h_a, dw, db, out);
}